// PteCriterion_2336462209676
// MI455X (gfx1250) — compile-verified
//
#include <hip/hip_runtime.h>
#include <hip/hip_bf16.h>

typedef __attribute__((ext_vector_type(2))) float v2f;
typedef __attribute__((ext_vector_type(8))) float v8f;

#define Bn 32
#define Sn 256
#define Vn 32000
#define Cn 100
#define Ln 8
#define CL_PAD 896     // 7 n-tiles * 128 K-values
#define C_PAD 112      // 7 n-tiles * 16 classes
#define NTILES 7
#define MTILES 2
#define THREADS 1024

__global__ __launch_bounds__(THREADS)
void pte_criterion_kernel(const float* __restrict__ logits,
                          const int* __restrict__ mlm_labels,
                          const int* __restrict__ m2c,
                          const float* __restrict__ filler_len,
                          const float* __restrict__ weight,
                          int* __restrict__ out)
{
    __shared__ float Xlds[Bn][CL_PAD];    // gathered+masked logits, zero-padded
    __shared__ float wnlds[CL_PAD];       // weight / filler_len, zero-padded
    __shared__ float clslds[Bn][C_PAD];   // class scores
    __shared__ int   poslds[Bn];

    const int tid = threadIdx.x;

    if (tid == 0) {
        __builtin_prefetch(m2c, 0, 0);        // global_prefetch_b8
        __builtin_prefetch(weight, 0, 0);
    }
    if (tid < Bn) poslds[tid] = Sn;
    __syncthreads();

    // ---- Phase A: first position with label >= 0 per batch ----
    for (int t = tid; t < Bn * Sn; t += THREADS) {
        if (mlm_labels[t] >= 0) atomicMin(&poslds[t / Sn], t % Sn);
    }
    // normalized weights: wn[c*8+l] = weight[c,l] / filler_len[c]
    for (int j = tid; j < CL_PAD; j += THREADS) {
        float wv = 0.0f;
        if (j < Cn * Ln) wv = weight[j] / filler_len[j >> 3];
        wnlds[j] = wv;
    }
    __syncthreads();

    // ---- Phase B: gather masked logits into LDS (mask m2c > 0) ----
    for (int t = tid; t < Bn * CL_PAD; t += THREADS) {
        const int b = t / CL_PAD;
        const int j = t % CL_PAD;
        float v = 0.0f;
        if (j < Cn * Ln) {
            const int idx = m2c[j];
            if (idx > 0) {
                const long off = ((long)b * Sn + poslds[b]) * (long)Vn + idx;
                v = logits[off];
            }
        }
        Xlds[b][j] = v;
    }
    __syncthreads();

    // ---- Phase C: cls(32x112) = X(32x896) * blockdiag-W(896x112) via WMMA f32 16x16x4 ----
    const int w    = tid >> 5;
    const int lane = tid & 31;
    if (w < MTILES * NTILES) {
        const int mt    = w / NTILES;
        const int nt    = w % NTILES;
        const int m     = lane & 15;   // A row / B-C-D column index
        const int h     = lane >> 4;   // lane half selects K pair / row half
        const int mrow  = mt * 16 + m;
        const int kbase = nt * 128;

        v8f acc = {0.f, 0.f, 0.f, 0.f, 0.f, 0.f, 0.f, 0.f};
        #pragma unroll 8
        for (int kk = 0; kk < 32; ++kk) {
            const int kl = kk * 4 + 2 * h;   // local K index in [0,128), always even
            v2f a, bf;
            a.x = Xlds[mrow][kbase + kl];
            a.y = Xlds[mrow][kbase + kl + 1];
            // unconditional LDS loads, then value-select (v_cndmask, no EXEC churn)
            const float w0 = wnlds[kbase + kl];
            const float w1 = wnlds[kbase + kl + 1];
            // block-diagonal: rows k in [8n, 8n+8) feed only class n; since kl is
            // even, both (kl>>3)==m and ((kl+1)>>3)==m reduce to (kk>>1)==m.
            const bool sel = ((kk >> 1) == m);
            bf.x = sel ? w0 : 0.f;
            bf.y = sel ? w1 : 0.f;
            acc = __builtin_amdgcn_wmma_f32_16x16x4_f32(
                false, a, false, bf, (short)0, acc, false, false);
        }
        #pragma unroll
        for (int r = 0; r < 8; ++r) {
            clslds[mt * 16 + r + 8 * h][nt * 16 + m] = acc[r];
        }
    }
    __syncthreads();

    // ---- Phase D: argmax over 100 classes (first-max semantics) ----
    if (tid < Bn) {
        float best  = clslds[tid][0];
        int   besti = 0;
        for (int c = 1; c < Cn; ++c) {
            const float v = clslds[tid][c];
            if (v > best) { best = v; besti = c; }
        }
        out[tid] = besti;
    }
}

extern "C" void kernel_launch(void* const* d_in, const int* in_sizes, int n_in,
                              void* d_out, int out_size, void* d_ws, size_t ws_size,
                              hipStream_t stream) {
    (void)in_sizes; (void)n_in; (void)out_size; (void)d_ws; (void)ws_size;
    const float* logits     = (const float*)d_in[0];
    const int*   mlm_labels = (const int*)d_in[1];
    const int*   m2c        = (const int*)d_in[2];
    const float* filler_len = (const float*)d_in[3];
    const float* weight     = (const float*)d_in[4];
    int* out = (int*)d_out;

    pte_criterion_kernel<<<dim3(1), dim3(THREADS), 0, stream>>>(
        logits, mlm_labels, m2c, filler_len, weight, out);
}